// MultislicePropagation_39230231281733
// MI455X (gfx1250) — compile-verified
//
#include <hip/hip_runtime.h>

// ---------------------------------------------------------------------------
// Multislice angular-spectrum propagation, 512x512 field, 100 slices.
// 1D 512-pt FFT done as matmul (512 = 16 x 32) on V_WMMA_F32_16X16X4_F32.
// Each 2D FFT = two 1D passes, each pass writes its output transposed via an
// LDS-staged tile so global stores are coalesced 128-bit segments.
// DFT matrices + twiddles precomputed once in global memory; each block copies
// them into LDS with GLOBAL_LOAD_ASYNC_TO_LDS (ASYNCcnt) when available.
// ---------------------------------------------------------------------------

typedef float v2f __attribute__((ext_vector_type(2)));
typedef float v8f __attribute__((ext_vector_type(8)));
typedef int   v4i __attribute__((vector_size(4 * sizeof(int))));
typedef __attribute__((address_space(1))) v4i* gptr_v4i;   // global int4*
typedef __attribute__((address_space(3))) v4i* lptr_v4i;   // LDS int4*

#define NYD 512
#define NXD 512
#define NZD 100
#define PS  0.1f
#define TWO_PI 6.28318530717958647692f
#define TABF 3584                      // floats per direction table (14 KB)

#if defined(__has_builtin)
#if __has_builtin(__builtin_amdgcn_global_load_async_to_lds_b128)
#define ASYNC_LDS 1
#endif
#endif

static __device__ inline v8f wmma_f32_16x16x4(v2f a, v2f b, v8f c) {
    return __builtin_amdgcn_wmma_f32_16x16x4_f32(false, a, false, b, (short)0, c,
                                                 false, false);
}

// ---------------------------------------------------------------------------
// Angular-spectrum kernels: kfwd = exp(+dz*phase), klast = conj(exp(+D*phase)),
// phase = i*2*pi*sqrt((n/l)^2 - fy^2 - fx^2); negative arg -> evanescent decay.
// ---------------------------------------------------------------------------
__global__ __launch_bounds__(256) void build_kernels(float* __restrict__ kf,
                                                     float* __restrict__ kl) {
    int idx = blockIdx.x * blockDim.x + threadIdx.x;
    if (idx >= NYD * NXD) return;
    int y = idx / NXD, x = idx % NXD;
    float fy = (float)(y < NYD / 2 ? y : y - NYD) / (NYD * PS);
    float fx = (float)(x < NXD / 2 ? x : x - NXD) / (NXD * PS);
    const float nl = 1.0f / 0.532f;
    const float dist = (NZD * 0.5f - 0.5f) * PS;    // 4.95
    float arg = nl * nl - fy * fy - fx * fx;
    if (arg >= 0.0f) {
        float s = sqrtf(arg);
        float sn, cs;
        __sincosf(TWO_PI * s * PS, &sn, &cs);
        kf[2 * idx] = cs; kf[2 * idx + 1] = sn;
        __sincosf(TWO_PI * s * dist, &sn, &cs);
        kl[2 * idx] = cs; kl[2 * idx + 1] = -sn;    // conj
    } else {
        float s = sqrtf(-arg);
        kf[2 * idx] = __expf(-TWO_PI * s * PS);   kf[2 * idx + 1] = 0.0f;
        kl[2 * idx] = __expf(-TWO_PI * s * dist); kl[2 * idx + 1] = 0.0f;
    }
}

// ---------------------------------------------------------------------------
// DFT matrix / twiddle tables for both directions.
// Layout per direction (floats): F16r[256] F16i[256] F32r[1024] F32i[1024]
//                                TWr[512] TWi[512]  == 3584 floats.
// 1792 complex entries; thread e computes one (cos,sin), writes fwd and inv.
// ---------------------------------------------------------------------------
__global__ __launch_bounds__(256) void build_tables(float* __restrict__ tabs) {
    int e = blockIdx.x * blockDim.x + threadIdx.x;
    if (e >= 1792) return;
    float* tf = tabs;                 // forward: exp(-i theta)
    float* ti = tabs + TABF;          // inverse: exp(+i theta)
    float th;
    int ro, io, idx;
    if (e < 256) {                    // F16: 16x16
        int r = e >> 4, c = e & 15;
        th = TWO_PI * (float)((r * c) & 15) / 16.0f;
        ro = 0; io = 256; idx = e;
    } else if (e < 1280) {            // F32: 32x32
        idx = e - 256;
        int r = idx >> 5, c = idx & 31;
        th = TWO_PI * (float)((r * c) & 31) / 32.0f;
        ro = 512; io = 1536;
    } else {                          // TW: W512^{k1*n2}, k1<16, n2<32
        idx = e - 1280;
        int r = idx >> 5, c = idx & 31;
        th = TWO_PI * (float)(r * c) / 512.0f;
        ro = 2560; io = 3072;
    }
    float sn, cs; __sincosf(th, &sn, &cs);
    tf[ro + idx] = cs; tf[io + idx] = -sn;
    ti[ro + idx] = cs; ti[io + idx] = sn;
}

// ---------------------------------------------------------------------------
// One 1D-FFT pass over all 512 rows; writes output TRANSPOSED out[k][row].
//   mulMode 0: plain   1: in *= obj[:,:,z]   2: in *= mul (spectral kernel)
// gtab = direction-specific table; scale applied at store (1/512 on inverse).
// One wave per row, 8 rows per 256-thread block.
// ---------------------------------------------------------------------------
__global__ __launch_bounds__(256) void fft_pass(const float* __restrict__ in,
                                                float* __restrict__ out,
                                                const float* __restrict__ mul,
                                                int mulMode, int z,
                                                const float* __restrict__ gtab,
                                                float scale) {
    __shared__ float tab[TABF];            // 14 KB: DFT matrices + twiddles
    __shared__ float smem[8 * 512 * 2];    // 32 KB: row scratch / output tile

    const int tid  = threadIdx.x;
    const int w    = tid >> 5;
    const int lane = tid & 31;
    const int lo   = lane & 15;
    const int hi   = lane >> 4;
    const int row0 = blockIdx.x * 8;
    const int row  = row0 + w;

    const float* F16r = tab;
    const float* F16i = tab + 256;
    const float* F32r = tab + 512;
    const float* F32i = tab + 1536;
    const float* TWr  = tab + 2560;
    const float* TWi  = tab + 3072;
    float* rowR = smem + w * 512;          // this wave's row (re)
    float* rowI = smem + 4096 + w * 512;   // this wave's row (im)

    // --- stage constant tables into LDS ------------------------------------
#ifdef ASYNC_LDS
    for (int j = tid; j < TABF / 4; j += 256) {     // 896 x b128 async copies
        __builtin_amdgcn_global_load_async_to_lds_b128(
            (gptr_v4i)(gtab + 4 * j), (lptr_v4i)(tab + 4 * j), 0, 0);
    }
#if __has_builtin(__builtin_amdgcn_s_wait_asynccnt)
    __builtin_amdgcn_s_wait_asynccnt(0);
#else
    asm volatile("s_wait_asynccnt 0x0" ::: "memory");
#endif
#else
    for (int j = tid; j < TABF / 4; j += 256)
        ((float4*)tab)[j] = ((const float4*)gtab)[j];
#endif
    __syncthreads();

    // --- load row (fused elementwise multiply) ------------------------------
    #pragma unroll
    for (int i = 0; i < 16; ++i) {
        int n = lane + 32 * i;
        int g = row * 512 + n;
        float2 v = ((const float2*)in)[g];
        if (mulMode == 1) {                 // obj slice z: [NY,NX,NZ,2]
            float2 o = ((const float2*)mul)[g * NZD + z];
            float re = v.x * o.x - v.y * o.y;
            float im = v.x * o.y + v.y * o.x;
            v.x = re; v.y = im;
        } else if (mulMode == 2) {          // spectral kernel: [NY,NX,2]
            float2 o = ((const float2*)mul)[g];
            float re = v.x * o.x - v.y * o.y;
            float im = v.x * o.y + v.y * o.x;
            v.x = re; v.y = im;
        }
        rowR[n] = v.x;                      // X^[n1][n2] row-major == linear n
        rowI[n] = v.y;
    }
    __syncthreads();

    // --- stage 1: Y(16x32) = F16(16x16) x X^(16x32), complex ----------------
    v8f yr[2], yi[2];
    yr[0] = (v8f)0.0f; yr[1] = (v8f)0.0f; yi[0] = (v8f)0.0f; yi[1] = (v8f)0.0f;
    #pragma unroll
    for (int ks = 0; ks < 4; ++ks) {
        int k0 = 4 * ks + 2 * hi;
        v2f ar, ai;
        ar.x = F16r[lo * 16 + k0];  ar.y = F16r[lo * 16 + k0 + 1];
        ai.x = F16i[lo * 16 + k0];  ai.y = F16i[lo * 16 + k0 + 1];
        v2f nai = -ai;
        #pragma unroll
        for (int t = 0; t < 2; ++t) {
            int col = 16 * t + lo;
            v2f br, bi;
            br.x = rowR[k0 * 32 + col];  br.y = rowR[(k0 + 1) * 32 + col];
            bi.x = rowI[k0 * 32 + col];  bi.y = rowI[(k0 + 1) * 32 + col];
            yr[t] = wmma_f32_16x16x4(ar, br, yr[t]);
            yr[t] = wmma_f32_16x16x4(nai, bi, yr[t]);
            yi[t] = wmma_f32_16x16x4(ar, bi, yi[t]);
            yi[t] = wmma_f32_16x16x4(ai, br, yi[t]);
        }
    }
    __syncthreads();                        // before overwriting row scratch

    // --- twiddle multiply, store T back to LDS ------------------------------
    #pragma unroll
    for (int t = 0; t < 2; ++t) {
        #pragma unroll
        for (int v = 0; v < 8; ++v) {
            int m = v + 8 * hi;             // k1
            int n2 = 16 * t + lo;           // n2
            float tr = TWr[m * 32 + n2], ti = TWi[m * 32 + n2];
            float a = yr[t][v], b = yi[t][v];
            rowR[m * 32 + n2] = a * tr - b * ti;
            rowI[m * 32 + n2] = a * ti + b * tr;
        }
    }
    __syncthreads();

    // --- stage 2: Z(16x32) = T(16x32) x F32(32x32), complex -----------------
    v8f zr[2], zi[2];
    zr[0] = (v8f)0.0f; zr[1] = (v8f)0.0f; zi[0] = (v8f)0.0f; zi[1] = (v8f)0.0f;
    #pragma unroll
    for (int ks = 0; ks < 8; ++ks) {
        int k0 = 4 * ks + 2 * hi;
        v2f ar, ai;
        ar.x = rowR[lo * 32 + k0];  ar.y = rowR[lo * 32 + k0 + 1];
        ai.x = rowI[lo * 32 + k0];  ai.y = rowI[lo * 32 + k0 + 1];
        v2f nai = -ai;
        #pragma unroll
        for (int t = 0; t < 2; ++t) {
            int col = 16 * t + lo;
            v2f br, bi;
            br.x = F32r[k0 * 32 + col];  br.y = F32r[(k0 + 1) * 32 + col];
            bi.x = F32i[k0 * 32 + col];  bi.y = F32i[(k0 + 1) * 32 + col];
            zr[t] = wmma_f32_16x16x4(ar, br, zr[t]);
            zr[t] = wmma_f32_16x16x4(nai, bi, zr[t]);
            zi[t] = wmma_f32_16x16x4(ar, bi, zi[t]);
            zi[t] = wmma_f32_16x16x4(ai, br, zi[t]);
        }
    }
    __syncthreads();                        // smem now reused as output tile

    // --- stage Z into LDS tile: tile[k][w], k = k1 + 16*k2 ------------------
    float2* tile = (float2*)smem;           // [512][8] float2 = 32 KB
    #pragma unroll
    for (int t = 0; t < 2; ++t) {
        #pragma unroll
        for (int v = 0; v < 8; ++v) {
            int k1 = v + 8 * hi;
            int k2 = 16 * t + lo;
            int k  = k1 + 16 * k2;
            float2 o;
            o.x = zr[t][v] * scale;
            o.y = zi[t][v] * scale;
            tile[k * 8 + w] = o;
        }
    }
    __syncthreads();

    // --- coalesced transposed store: out[k*512 + row0 .. +7] (64B segs) -----
    for (int k = tid; k < 512; k += 256) {
        const float4* src = (const float4*)(tile + k * 8);  // 8 float2 = 4xf4
        float4* dst = ((float4*)out) + (size_t)(k * 512 + row0) / 2;
        dst[0] = src[0]; dst[1] = src[1]; dst[2] = src[2]; dst[3] = src[3];
    }
}

// ---------------------------------------------------------------------------
extern "C" void kernel_launch(void* const* d_in, const int* in_sizes, int n_in,
                              void* d_out, int out_size, void* d_ws, size_t ws_size,
                              hipStream_t stream) {
    const float* obj      = (const float*)d_in[0];   // [512,512,100,2] f32
    const float* field_in = (const float*)d_in[1];   // [512,512,2]     f32
    float* outp = (float*)d_out;                     // [512,512,2]     f32

    const size_t planeF = (size_t)NYD * NXD * 2;     // floats per complex plane
    float* ws   = (float*)d_ws;
    float* kf   = ws;                     // kernel_fwd
    float* kl   = ws + planeF;            // kernel_last
    float* X    = ws + 2 * planeF;        // current field
    float* T1   = ws + 3 * planeF;        // ping
    float* T2   = ws + 4 * planeF;        // pong
    float* tabs = ws + 5 * planeF;        // [2][3584] DFT tables (fwd, inv)
    float* tabF = tabs;
    float* tabI = tabs + TABF;

    build_kernels<<<(NYD * NXD + 255) / 256, 256, 0, stream>>>(kf, kl);
    build_tables<<<7, 256, 0, stream>>>(tabs);
    (void)hipMemcpyAsync(X, field_in, planeF * sizeof(float),
                         hipMemcpyDeviceToDevice, stream);

    const float invN = 1.0f / 512.0f;
    for (int z = 0; z < NZD; ++z) {
        const float* K = (z == NZD - 1) ? kl : kf;
        float* dst     = (z == NZD - 1) ? outp : X;
        // forward FFT2: rows then cols (each pass writes transposed)
        fft_pass<<<64, 256, 0, stream>>>(X,  T1, obj, 1, z, tabF, 1.0f);
        fft_pass<<<64, 256, 0, stream>>>(T1, T2, nullptr, 0, 0, tabF, 1.0f);
        // spectral multiply fused into first inverse pass; inverse FFT2
        fft_pass<<<64, 256, 0, stream>>>(T2, T1, K, 2, 0, tabI, invN);
        fft_pass<<<64, 256, 0, stream>>>(T1, dst, nullptr, 0, 0, tabI, invN);
    }
}